// MolecularEncoder_90589450207900
// MI455X (gfx1250) — compile-verified
//
#include <hip/hip_runtime.h>

typedef __attribute__((ext_vector_type(2))) float v2f;
typedef __attribute__((ext_vector_type(8))) float v8f;

#define HID 64
#define CDIV(a, b) (((a) + (b) - 1) / (b))

// ---------------- degree / norm ----------------
__global__ void k_init_deg(float* deg, int n) {
  int i = blockIdx.x * blockDim.x + threadIdx.x;
  if (i < n) deg[i] = 1.0f;  // self-loop contributes 1 to every node
}
__global__ void k_accum_deg(const int* __restrict__ dst, float* deg, int e) {
  int i = blockIdx.x * blockDim.x + threadIdx.x;
  if (i < e) atomicAdd(&deg[dst[i]], 1.0f);
}
__global__ void k_dinv(float* deg, int n) {
  int i = blockIdx.x * blockDim.x + threadIdx.x;
  if (i < n) deg[i] = rsqrtf(deg[i]);  // deg >= 1 always
}

// ---------------- input GEMM (K=5, too skinny for WMMA) ----------------
__global__ void k_in_gemm(const float* __restrict__ x, const float* __restrict__ W1,
                          float* __restrict__ h, int n) {
  int idx = blockIdx.x * blockDim.x + threadIdx.x;
  if (idx >= n * HID) return;
  int node = idx >> 6, c = idx & 63;
  const float* xr = x + node * 5;
  float s = 0.0f;
#pragma unroll
  for (int k = 0; k < 5; ++k) s = fmaf(xr[k], W1[k * HID + c], s);
  h[idx] = s;  // bias is applied post-aggregation (folded into agg init)
}

// ---------------- aggregation ----------------
__global__ void k_init_bias(float* agg, const float* __restrict__ b, int total) {
  int idx = blockIdx.x * blockDim.x + threadIdx.x;
  if (idx < total) agg[idx] = b[idx & 63];
}
__global__ void k_scatter(const int* __restrict__ src, const int* __restrict__ dst,
                          const float* __restrict__ dinv, const float* __restrict__ h,
                          float* agg, int E) {
  int t = blockIdx.x * blockDim.x + threadIdx.x;
  int e = t >> 6;
  if (e >= E) return;
  int c = t & 63;
  int s = src[e], d = dst[e];
  float w = dinv[s] * dinv[d];
  atomicAdd(&agg[(size_t)d * HID + c], h[(size_t)s * HID + c] * w);
}
__global__ void k_self(const float* __restrict__ dinv, const float* __restrict__ h,
                       float* agg, int n) {
  int idx = blockIdx.x * blockDim.x + threadIdx.x;
  if (idx >= n * HID) return;
  float w = dinv[idx >> 6];
  agg[idx] += h[idx] * (w * w);
}
__global__ void k_relu(float* a, int total) {
  int idx = blockIdx.x * blockDim.x + threadIdx.x;
  if (idx < total) a[idx] = fmaxf(a[idx], 0.0f);
}

// ---------------- WMMA f32 GEMM: C[M,NC] = A[M,64] x B[64,NC] (+bias)(+relu) ----------------
// One wave owns a 16-row tile and all NC columns. Uses V_WMMA_F32_16X16X4_F32
// (f32 A/B/C -> exact numerics). B staged in LDS. EXEC stays all-ones through
// the WMMAs (row clamp instead of predication), as required by the ISA.
template <int NT, bool BIAS, bool RELU>
__global__ __launch_bounds__(128) void k_wmma_gemm(const float* __restrict__ A,
                                                   const float* __restrict__ B,
                                                   const float* __restrict__ bias,
                                                   float* __restrict__ C, int M) {
  constexpr int K = 64;
  constexpr int NC = NT * 16;
  __shared__ float sB[K * NC];
  for (int i = threadIdx.x; i < K * NC; i += blockDim.x) sB[i] = B[i];
  __syncthreads();

  const int lane = threadIdx.x & 31;
  const int wave = threadIdx.x >> 5;
  const int tile = blockIdx.x * (blockDim.x >> 5) + wave;
  const int row0 = tile * 16;
  if (row0 >= M) return;  // wave-uniform exit

  const int l15 = lane & 15;
  const int hi  = lane >> 4;  // 0: K offsets 0,1 ; 1: K offsets 2,3 (ISA 16x4 A layout)

  v8f acc[NT];
  v8f zero = {0.f, 0.f, 0.f, 0.f, 0.f, 0.f, 0.f, 0.f};
#pragma unroll
  for (int i = 0; i < NT; ++i) acc[i] = zero;

  int rowA = row0 + l15;
  if (rowA >= M) rowA = M - 1;  // clamp: keep all lanes live for WMMA
  const float* Arow = A + (size_t)rowA * K;

  for (int k0 = 0; k0 < K; k0 += 4) {
    const int ka = k0 + hi * 2;
    v2f a;
    a.x = Arow[ka];
    a.y = Arow[ka + 1];
#pragma unroll
    for (int nt = 0; nt < NT; ++nt) {
      const int col = nt * 16 + l15;
      v2f b;
      b.x = sB[ka * NC + col];
      b.y = sB[(ka + 1) * NC + col];
      acc[nt] = __builtin_amdgcn_wmma_f32_16x16x4_f32(
          /*neg_a=*/false, a, /*neg_b=*/false, b,
          /*c_mod=*/(short)0, acc[nt], /*reuse_a=*/false, /*reuse_b=*/false);
    }
  }

  // D layout: VGPR j -> row j (lanes 0-15) / row j+8 (lanes 16-31), col = lane&15
  const int rbase = row0 + hi * 8;
#pragma unroll
  for (int nt = 0; nt < NT; ++nt) {
    const int col = nt * 16 + l15;
    float bv = BIAS ? bias[col] : 0.0f;
#pragma unroll
    for (int j = 0; j < 8; ++j) {
      int r = rbase + j;
      if (r < M) {
        float v = acc[nt][j] + bv;
        if (RELU) v = fmaxf(v, 0.0f);
        C[(size_t)r * NC + col] = v;
      }
    }
  }
}

// ---------------- pooling ----------------
__global__ void k_zero(float* p, int total) {
  int idx = blockIdx.x * blockDim.x + threadIdx.x;
  if (idx < total) p[idx] = 0.0f;
}
__global__ void k_pool(const int* __restrict__ batch, const float* __restrict__ h,
                       float* pooled, float* cnt, int n) {
  int t = blockIdx.x * blockDim.x + threadIdx.x;
  int node = t >> 6;
  if (node >= n) return;
  int c = t & 63;
  int g = batch[node];
  atomicAdd(&pooled[(size_t)g * HID + c], h[(size_t)node * HID + c]);
  if (c == 0) atomicAdd(&cnt[g], 1.0f);
}
__global__ void k_pooldiv(float* pooled, const float* __restrict__ cnt, int total) {
  int idx = blockIdx.x * blockDim.x + threadIdx.x;
  if (idx < total) pooled[idx] /= fmaxf(cnt[idx >> 6], 1.0f);
}

// ---------------- driver ----------------
extern "C" void kernel_launch(void* const* d_in, const int* in_sizes, int n_in,
                              void* d_out, int out_size, void* d_ws, size_t ws_size,
                              hipStream_t stream) {
  (void)n_in; (void)ws_size;
  const float* x     = (const float*)d_in[0];
  const int*   ei    = (const int*)d_in[1];   // JAX default: int64 demotes to int32
  const int*   batch = (const int*)d_in[2];
  const float* W1 = (const float*)d_in[3];
  const float* b1 = (const float*)d_in[4];
  const float* W2 = (const float*)d_in[5];
  const float* b2 = (const float*)d_in[6];
  const float* W3 = (const float*)d_in[7];
  const float* b3 = (const float*)d_in[8];
  const float* W4 = (const float*)d_in[9];
  const float* b4 = (const float*)d_in[10];
  float* out = (float*)d_out;

  const int N = in_sizes[0] / 5;
  const int E = in_sizes[1] / 2;
  const int G = out_size / 128;
  const int* src = ei;
  const int* dst = ei + E;

  char* w = (char*)d_ws;
  size_t o = 0;
  auto take = [&](size_t bytes) -> char* {
    char* p = w + o;
    o += (bytes + 255) & ~(size_t)255;
    return p;
  };
  float* dinv   = (float*)take((size_t)N * 4);
  float* hbuf   = (float*)take((size_t)N * HID * 4);
  float* agg    = (float*)take((size_t)N * HID * 4);
  float* pooled = (float*)take((size_t)G * HID * 4);
  float* cnt    = (float*)take((size_t)G * 4);
  float* z      = (float*)take((size_t)G * HID * 4);

  const int NH = N * HID;             // 6.4M
  const long long ETH = (long long)E * HID;  // 64M scatter threads
  const int sblk = (int)((ETH + 255) / 256);
  const int wblkN = CDIV(CDIV(N, 16), 4);  // waves of 16 rows, 4 waves/block
  const int wblkG = CDIV(CDIV(G, 16), 4);

  // degrees -> dinv
  k_init_deg<<<CDIV(N, 256), 256, 0, stream>>>(dinv, N);
  k_accum_deg<<<CDIV(E, 256), 256, 0, stream>>>(dst, dinv, E);
  k_dinv<<<CDIV(N, 256), 256, 0, stream>>>(dinv, N);

  // layer 1: h = x @ W1 ; agg = scatter(norm * h[src]) + b1 ; relu
  k_in_gemm<<<CDIV(NH, 256), 256, 0, stream>>>(x, W1, hbuf, N);
  k_init_bias<<<CDIV(NH, 256), 256, 0, stream>>>(agg, b1, NH);
  k_scatter<<<sblk, 256, 0, stream>>>(src, dst, dinv, hbuf, agg, E);
  k_self<<<CDIV(NH, 256), 256, 0, stream>>>(dinv, hbuf, agg, N);
  k_relu<<<CDIV(NH, 256), 256, 0, stream>>>(agg, NH);

  // layer 2: h = agg @ W2 (WMMA) ; agg = scatter + b2 ; relu
  k_wmma_gemm<4, false, false><<<wblkN, 128, 0, stream>>>(agg, W2, nullptr, hbuf, N);
  k_init_bias<<<CDIV(NH, 256), 256, 0, stream>>>(agg, b2, NH);
  k_scatter<<<sblk, 256, 0, stream>>>(src, dst, dinv, hbuf, agg, E);
  k_self<<<CDIV(NH, 256), 256, 0, stream>>>(dinv, hbuf, agg, N);
  k_relu<<<CDIV(NH, 256), 256, 0, stream>>>(agg, NH);

  // global mean pool
  k_zero<<<CDIV(G * HID, 256), 256, 0, stream>>>(pooled, G * HID);
  k_zero<<<CDIV(G, 256), 256, 0, stream>>>(cnt, G);
  k_pool<<<CDIV(NH, 256), 256, 0, stream>>>(batch, agg, pooled, cnt, N);
  k_pooldiv<<<CDIV(G * HID, 256), 256, 0, stream>>>(pooled, cnt, G * HID);

  // MLP head (WMMA): z = relu(pooled@W3 + b3) ; out = z@W4 + b4
  k_wmma_gemm<4, true, true><<<wblkG, 128, 0, stream>>>(pooled, W3, b3, z, G);
  k_wmma_gemm<8, true, false><<<wblkG, 128, 0, stream>>>(z, W4, b4, out, G);
}